// PathBasedLoss_74096775790628
// MI455X (gfx1250) — compile-verified
//
#include <hip/hip_runtime.h>

#define PBL_BLOCK 256
#define WPB 8                     // wave32 waves per block
#define TILE_ROWS 32              // one row (8 floats) per lane per tile
#define TILE_BYTES 1024           // 32 rows * 32 B

typedef __attribute__((address_space(3))) char lds_char_t;

// Zero the scratch accumulator (on-stream, graph-capture safe).
__global__ __launch_bounds__(PBL_BLOCK) void pbl_zero(unsigned int* __restrict__ cnt) {
  if (blockIdx.x == 0 && threadIdx.x == 0) cnt[0] = 0u;
}

// Issue one 1 KiB tile (32 rows x 32 B) as two async b128 copies into LDS.
// Lane L moves contiguous bytes [L*16, L*16+16) and [512+L*16, ...): each issue
// is a fully coalesced 512 B transfer; the copy is an identity byte layout.
__device__ __forceinline__ void async_tile_to_lds(const char* gbase,
                                                  lds_char_t* lbase, int lane) {
  asm volatile("global_load_async_to_lds_b128 %0, %1, off"
               :: "v"(lbase + lane * 16), "v"(gbase + lane * 16)
               : "memory");
  asm volatile("global_load_async_to_lds_b128 %0, %1, off"
               :: "v"(lbase + 512 + lane * 16), "v"(gbase + 512 + lane * 16)
               : "memory");
}

__device__ __forceinline__ unsigned int row_mismatch(float4 a, float4 b, int t) {
  // Group-of-argmax: per-group maxima; (>=) ordering preserves jnp.argmax
  // first-occurrence tie semantics because all class indices of an earlier
  // group precede those of a later group.
  float m0 = fmaxf(a.x, fmaxf(a.y, a.z));   // classes 0,1,2 -> group 0
  float m1 = fmaxf(a.w, fmaxf(b.x, b.y));   // classes 3,4,5 -> group 1
  float m2 = fmaxf(b.z, b.w);               // classes 6,7   -> group 2
  int pg = (m0 >= m1) ? ((m0 >= m2) ? 0 : 2) : ((m1 >= m2) ? 1 : 2);
  int tg = (t > 2) + (t > 5);
  return (unsigned int)(pg != tg);
}

// Streaming mismatch counter with per-wave double-buffered async-to-LDS
// staging of the prediction stream. Each wave owns its LDS region, so no
// block barriers: s_wait_asynccnt orders producer->consumer, and an explicit
// s_wait_dscnt 0 after the LDS reads orders consumer->next-producer (WAR on
// the buffer being refilled next iteration).
__global__ __launch_bounds__(PBL_BLOCK) void pbl_count(
    const float* __restrict__ preds,
    const int* __restrict__ tgt2,        // int64 targets viewed as LE dword pairs
    unsigned int* __restrict__ cnt,
    int nrows) {
  __shared__ float4 lbuf[WPB][2][TILE_BYTES / 16];   // 16 KiB
  __shared__ unsigned int sm[PBL_BLOCK];

  const int lane   = threadIdx.x & 31;
  const int wave   = threadIdx.x >> 5;
  const int ntiles = nrows / TILE_ROWS;
  const int tile0  = blockIdx.x * WPB + wave;
  const int tstr   = gridDim.x * WPB;
  const char* gpred = (const char*)preds;
  unsigned int c = 0;

  if (tile0 < ntiles) {
    async_tile_to_lds(gpred + (size_t)tile0 * TILE_BYTES,
                      (lds_char_t*)&lbuf[wave][0][0], lane);
  }
  int pb = 0;
  for (int t = tile0; t < ntiles; t += tstr) {
    const int nxt = t + tstr;
    if (nxt < ntiles) {
      async_tile_to_lds(gpred + (size_t)nxt * TILE_BYTES,
                        (lds_char_t*)&lbuf[wave][pb ^ 1][0], lane);
      __builtin_prefetch(&tgt2[2 * (nxt * TILE_ROWS + lane)], 0, 0);
      asm volatile("s_wait_asynccnt 0x2" ::: "memory");   // tile t's 2 ops done
    } else {
      asm volatile("s_wait_asynccnt 0x0" ::: "memory");
    }
    float4 a = lbuf[wave][pb][2 * lane + 0];   // ds_load_b128: this lane's row
    float4 b = lbuf[wave][pb][2 * lane + 1];
    // Pin the ds reads' completion BEFORE the next iteration's async issue
    // overwrites this buffer (airtight WAR ordering; near-zero cost since the
    // data is consumed by VALU immediately below anyway).
    asm volatile("s_wait_dscnt 0x0" ::: "memory");
    int tt = tgt2[2 * (t * TILE_ROWS + lane)]; // low dword of int64 target
    c += row_mismatch(a, b, tt);
    pb ^= 1;
  }

  // Tail rows (nrows % 32 != 0; zero for B = 4M) via direct loads, block 0 wave 0.
  if (blockIdx.x == 0 && wave == 0) {
    const float4* p4 = reinterpret_cast<const float4*>(preds);
    for (int r = ntiles * TILE_ROWS + lane; r < nrows; r += 32) {
      c += row_mismatch(p4[2 * r], p4[2 * r + 1], tgt2[2 * r]);
    }
  }

  // Block reduction, then one exact integer atomic per block.
  sm[threadIdx.x] = c;
  __syncthreads();
#pragma unroll
  for (int off = PBL_BLOCK / 2; off > 0; off >>= 1) {
    if ((int)threadIdx.x < off) sm[threadIdx.x] += sm[threadIdx.x + off];
    __syncthreads();
  }
  if (threadIdx.x == 0) atomicAdd(cnt, sm[0]);
}

// count / B  (count < 2^24 => exact in fp32; division by 2^22 is exact).
__global__ void pbl_final(const unsigned int* __restrict__ cnt,
                          float* __restrict__ out, float inv_n) {
  if (blockIdx.x == 0 && threadIdx.x == 0) out[0] = (float)cnt[0] * inv_n;
}

extern "C" void kernel_launch(void* const* d_in, const int* in_sizes, int n_in,
                              void* d_out, int out_size, void* d_ws, size_t ws_size,
                              hipStream_t stream) {
  const float*  preds = (const float*)d_in[0];     // [B, 8] float32
  const int*    tgt2  = (const int*)d_in[1];       // [B] int64, viewed as dword pairs
  float*        out   = (float*)d_out;             // scalar float32
  unsigned int* cnt   = (unsigned int*)d_ws;       // scratch accumulator

  const int nrows  = in_sizes[1];                  // B = 4194304
  const int blocks = 4096;                         // 32768 waves, 4 tiles each

  pbl_zero <<<1, PBL_BLOCK, 0, stream>>>(cnt);
  pbl_count<<<blocks, PBL_BLOCK, 0, stream>>>(preds, tgt2, cnt, nrows);
  pbl_final<<<1, 32, 0, stream>>>(cnt, out, 1.0f / (float)nrows);
}